// Conv_46265387713122
// MI455X (gfx1250) — compile-verified
//
#include <hip/hip_runtime.h>
#include <hip/hip_bf16.h>

// CDNA5 / gfx1250, wave32. Implicit-GEMM interval-bound conv:
//   V_WMMA_F32_16X16X4_F32 compute core (full fp32; workload is HBM-bound at
//   ~380 MB / 23.3 TB/s ~= 16 us, so no precision downgrade buys anything),
//   GLOBAL_LOAD_ASYNC_TO_LDS_B128 staging (coalesced 16B/lane, ASYNCcnt).

typedef float v2f __attribute__((ext_vector_type(2)));
typedef float v8f __attribute__((ext_vector_type(8)));

static constexpr int IN_H  = 1536;
static constexpr int IN_W  = 1536;
static constexpr int OUT_H = 768;
static constexpr int OUT_W = 768;
static constexpr int NCH   = 16;                 // C_in == C_out == 16
static constexpr int HWsz  = IN_H * IN_W;        // input plane (elems)
static constexpr int OHW   = OUT_H * OUT_W;      // output plane (elems)
static constexpr int NSTEP = 36;                 // K = 144 = 36 * 4
static constexpr int FRAG  = 64;                 // floats per A-fragment

// Input staging: per block (one oy, 128 ox), for each {l,u} x 16ch x 3 rows,
// one 272-float segment covering ix in [2*ox0-8, 2*ox0+263] (covers all
// ix = 2*ox+kw-1 needed, 16B-aligned at both ends).
static constexpr int SEGLEN   = 272;             // floats per segment
static constexpr int NCHUNK   = SEGLEN / 4;      // 68 x 16B chunks
static constexpr int NSEG     = 2 * NCH * 3;     // 96 segments
static constexpr int TOTCHUNK = NSEG * NCHUNK;   // 6528 = 25*256 + 128 (whole waves)
static constexpr size_t SMEM_BYTES = (size_t)NSEG * SEGLEN * 4;  // 104448 B

// Per-wave tile: D[16 oc x 16 ox] at one oy, both bounds.
// A (16x4 f32, M=oc): lane L holds M = L&15; VGPR v holds K = 2*(L>>4)+v.
// B (4x16 f32, N=ox): lane L holds N = L&15; VGPR v holds K = 2*(L>>4)+v.
// D (16x16 f32):      VGPR r, lane L -> M = r + 8*(L>>4), N = L&15.

__global__ __launch_bounds__(256) void ibp_conv3x3_s2_wmma(
    const float* __restrict__ l, const float* __restrict__ u,
    const float* __restrict__ weight, const float* __restrict__ bias,
    float* __restrict__ out)
{
    extern __shared__ float sIn[];               // staged input rows (dynamic)
    __shared__ float sW[2 * NSTEP * FRAG];       // w_pos | w_neg fragments

    const int oy    = blockIdx.y;
    const int ox0   = blockIdx.x * 128;
    const int xbase = 2 * ox0 - 8;               // segment origin in input x

    // ---- 1) Issue async global->LDS copies (ASYNCcnt-tracked, 16B/lane) ----
    #pragma unroll
    for (int k = 0; k < 7; ++k) {
        int i = threadIdx.x + k * 256;
        if (i < TOTCHUNK) {                      // tail is whole waves: no divergence
            int seg = i / NCHUNK;                // ((t*16 + c)*3 + r)
            int j   = i - seg * NCHUNK;
            int r   = seg % 3;
            int c   = (seg / 3) & 15;
            int t   = seg / 48;
            int iy  = 2 * oy + r - 1;
            int iyc = min(max(iy, 0), IN_H - 1);
            int xc  = xbase + 4 * j;
            int xcc = min(max(xc, 0), IN_W - 4); // fully-OOB chunks clamp; never
                                                 // consumed as valid data
            const float* src = (t ? u : l)
                + (size_t)c * HWsz + (size_t)iyc * IN_W + xcc;
            unsigned dst = (unsigned)(uintptr_t)&sIn[seg * SEGLEN + 4 * j];
            asm volatile("global_load_async_to_lds_b128 %0, %1, off"
                         :: "v"(dst), "v"(src) : "memory");
        }
    }

    // ---- 2) Stage weights split into pos/neg WMMA A-fragments -------------
    for (int idx = threadIdx.x; idx < NSTEP * FRAG; idx += 256) {
        int s    = idx >> 6;                     // k-step 0..35
        int rr   = idx & 63;
        int lane = rr >> 1;
        int v    = rr & 1;
        int oc   = lane & 15;
        int hv   = lane >> 4;
        int g    = s & 3;                        // ic group of 4
        int p    = s >> 2;                       // kh*3+kw
        int kh   = p / 3, kw = p % 3;
        int ic   = 4 * g + 2 * hv + v;
        float w  = weight[oc * (NCH * 9) + ic * 9 + kh * 3 + kw];
        sW[idx]                = fmaxf(w, 0.0f);
        sW[NSTEP * FRAG + idx] = fminf(w, 0.0f);
    }

    // ---- 3) Wait for async staging, then make LDS visible to all waves ----
    asm volatile("s_wait_asynccnt 0x0" ::: "memory");
    __syncthreads();

    // ---- 4) WMMA hot loop: all operands from LDS --------------------------
    const int lane = threadIdx.x & 31;
    const int wave = threadIdx.x >> 5;
    const int n    = lane & 15;                  // N index (ox within tile)
    const int hv   = lane >> 4;
    const int ox   = ox0 + wave * 16 + n;

    v8f accL = {0.f, 0.f, 0.f, 0.f, 0.f, 0.f, 0.f, 0.f};
    v8f accU = {0.f, 0.f, 0.f, 0.f, 0.f, 0.f, 0.f, 0.f};

    #pragma unroll
    for (int kh = 0; kh < 3; ++kh) {
        int  iy    = 2 * oy + kh - 1;
        bool rowok = (iy >= 0) && (iy < IN_H);
        #pragma unroll
        for (int kw = 0; kw < 3; ++kw) {
            int  ix  = 2 * ox + kw - 1;
            bool inb = rowok && (ix >= 0) && (ix < IN_W);
            int  pos = 32 * wave + 2 * n + kw + 7;   // == ix - xbase
            int  sbase = (kh * 3 + kw) * 4;
            #pragma unroll
            for (int g = 0; g < 4; ++g) {
                int s = sbase + g;
                v2f apos = *(const v2f*)&sW[s * FRAG + lane * 2];
                v2f aneg = *(const v2f*)&sW[NSTEP * FRAG + s * FRAG + lane * 2];

                int c0 = 4 * g + 2 * hv;         // this half-wave's K channels
                // l at segments [c*3+kh], u at +48 segments; channel delta =
                // 3*SEGLEN floats -> all fold into 16-bit DS offsets.
                const float* p0 = &sIn[(c0 * 3 + kh) * SEGLEN + pos];
                float bl0 = p0[0];
                float bl1 = p0[3 * SEGLEN];
                float bu0 = p0[48 * SEGLEN];
                float bu1 = p0[51 * SEGLEN];
                // zero-pad via value select (EXEC stays all-1s for WMMA)
                bl0 = inb ? bl0 : 0.0f;  bl1 = inb ? bl1 : 0.0f;
                bu0 = inb ? bu0 : 0.0f;  bu1 = inb ? bu1 : 0.0f;
                v2f bl = {bl0, bl1};
                v2f bu = {bu0, bu1};

                accL = __builtin_amdgcn_wmma_f32_16x16x4_f32(
                    false, apos, false, bl, (short)0, accL, false, false);
                accL = __builtin_amdgcn_wmma_f32_16x16x4_f32(
                    false, aneg, false, bu, (short)0, accL, false, false);
                accU = __builtin_amdgcn_wmma_f32_16x16x4_f32(
                    false, apos, false, bu, (short)0, accU, false, false);
                accU = __builtin_amdgcn_wmma_f32_16x16x4_f32(
                    false, aneg, false, bl, (short)0, accU, false, false);
            }
        }
    }

    // ---- 5) Bias + store. D row M = r + 8*hv is the output channel. -------
    float* __restrict__ outL = out;
    float* __restrict__ outU = out + (size_t)NCH * OHW;
    #pragma unroll
    for (int r = 0; r < 8; ++r) {
        int    oc = r + 8 * hv;
        float  b  = bias[oc];
        size_t o  = (size_t)oc * OHW + (size_t)oy * OUT_W + ox;
        outL[o] = accL[r] + b;   // lanes 0-15 / 16-31: two 64B coalesced runs
        outU[o] = accU[r] + b;
    }
}

extern "C" void kernel_launch(void* const* d_in, const int* in_sizes, int n_in,
                              void* d_out, int out_size, void* d_ws, size_t ws_size,
                              hipStream_t stream) {
    const float* l      = (const float*)d_in[0];
    const float* u      = (const float*)d_in[1];
    const float* weight = (const float*)d_in[2];
    const float* bias   = (const float*)d_in[3];
    float* out = (float*)d_out;

    // 768 ox = 6 blocks x 8 waves x 16; 768 oy rows. Exact tiling, no remainder.
    // Dynamic LDS: 102 KB staged input (+18 KB static weights) -> 120 KB/WG,
    // legal on CDNA5 (320 KB/WGP), 2 blocks resident per WGP.
    dim3 grid(OUT_W / 128, OUT_H, 1);
    dim3 block(256, 1, 1);
    ibp_conv3x3_s2_wmma<<<grid, block, SMEM_BYTES, stream>>>(l, u, weight, bias, out);
}